// ConvolutionalLUTLayer_47579647705178
// MI455X (gfx1250) — compile-verified
//
#include <hip/hip_runtime.h>
#include <math.h>

// ---------------------------------------------------------------------------
// Differentiable conv-LUT network on MI455X (gfx1250, wave32).
//
// N=1024 pixels (64 images x 16 positions), T=128 trees,
// layers of 36 -> 6 -> 1 soft-LUT nodes (6 inputs each, 64-entry LUT).
//
// Inner contraction per node maps to V_WMMA_F32_16X16X4_F32:
//   D[pq, n] = A[pq, r] * B[r, n],  A = sigmoid(lut)[t,m] (shared across
//   pixels), B[r, n] = Wc[pixel n][r] (per-pixel trilinear factor).
// Remaining q/p contraction done per-lane on the D layout + one shfl_xor(16).
// Lane halves (l, l^16) hold the same pixel, so after the shuffle both halves
// hold the identical node value and can store unconditionally (no exec mask).
// ---------------------------------------------------------------------------

typedef float v2f __attribute__((ext_vector_type(2)));
typedef float v8f __attribute__((ext_vector_type(8)));

static constexpr int kT  = 128;   // trees
static constexpr int kM0 = 36;    // layer0 nodes per tree
static constexpr int kM1 = 6;     // layer1 nodes per tree
static constexpr int kN0 = kT * kM0 * 64;   // 294912 sigmoid lut0 floats
static constexpr int kN1 = kT * kM1 * 64;   // 49152
static constexpr int kN2 = kT * 64;         // 8192

// ---- kernel 1: precompute sigmoid(lut*) into workspace (n-invariant) ------
__global__ void sigmoid_pre_kernel(const float* __restrict__ lut0,
                                   const float* __restrict__ lut1,
                                   const float* __restrict__ lut2,
                                   float* __restrict__ ws) {
  int i = blockIdx.x * blockDim.x + threadIdx.x;
  if (i < kN0) {
    ws[i] = 1.0f / (1.0f + expf(-lut0[i]));
  } else if (i < kN0 + kN1) {
    int j = i - kN0;
    ws[i] = 1.0f / (1.0f + expf(-lut1[j]));
  } else if (i < kN0 + kN1 + kN2) {
    int j = i - kN0 - kN1;
    ws[i] = 1.0f / (1.0f + expf(-lut2[j]));
  }
}

// ---- one soft-LUT node via WMMA -------------------------------------------
// g[0..5]: gathered inputs for this lane's pixel (lanes l and l^16 share a
// pixel). sigrow: 64 precomputed sigmoid entries for (t, m).
// Layouts (32-bit, wave32):
//   A 16x4 : lane = row pq;     V0/V1 = K (r) = {0,1} (lanes<16) or {2,3}.
//   B 4x16 : lane = col n(&15); V0/V1 = K (r) = {0,1} (lanes<16) or {2,3}.
//   D 16x16: lane = col n(&15); V0..7 = rows pq 0..7 (lanes<16) or 8..15.
__device__ __forceinline__ float lut_node(const float g[6],
                                          const float* __restrict__ sigrow,
                                          int lane) {
  const bool hi = lane >= 16;

  // B operand: Wc = pair_w(g4, g5) = [(1-a)(1-b), (1-a)b, a(1-b), ab]
  float ca = hi ? g[4] : (1.0f - g[4]);
  v2f Bv;
  Bv.x = ca * (1.0f - g[5]);
  Bv.y = ca * g[5];

  // A operand: sigmoid(lut)[pq*4 + r], r block selected by lane half
  int pq = lane & 15;
  const float* ap = sigrow + pq * 4 + (hi ? 2 : 0);
  v2f Av;
  Av.x = ap[0];
  Av.y = ap[1];

  v8f C = {};
  // D = A(16x4) x B(4x16) + 0   -> t1[pq, n]
  v8f D = __builtin_amdgcn_wmma_f32_16x16x4_f32(
      /*neg_a=*/false, Av, /*neg_b=*/false, Bv,
      /*c_mod=*/(short)0, C, /*reuse_a=*/false, /*reuse_b=*/false);

  // q/p contraction on this lane's 8 rows:
  // row j (lanes<16):  pq = j     -> pa = j>>2 in {0,1}, qb = j&3
  // row j (lanes>=16): pq = 8 + j -> pa = 2 + (j>>2),    qb = j&3
  float wb0 = (1.0f - g[2]) * (1.0f - g[3]);
  float wb1 = (1.0f - g[2]) * g[3];
  float wb2 = g[2] * (1.0f - g[3]);
  float wb3 = g[2] * g[3];
  float asel = hi ? g[0] : (1.0f - g[0]);
  float wlo = asel * (1.0f - g[1]);   // pa even within this half
  float whi = asel * g[1];            // pa odd  within this half

  float s_lo = D[0] * wb0 + D[1] * wb1 + D[2] * wb2 + D[3] * wb3;
  float s_hi = D[4] * wb0 + D[5] * wb1 + D[6] * wb2 + D[7] * wb3;
  float partial = wlo * s_lo + whi * s_hi;

  // lane l holds pq 0..7, lane l^16 holds pq 8..15 of the same pixel
  return partial + __shfl_xor(partial, 16, 32);
}

// ---- kernel 2: one wave32 block per (tree, image) -------------------------
__global__ __launch_bounds__(32)
void lutconv_kernel(const float* __restrict__ x,     // (64,1,8,8)
                    const int*   __restrict__ idx0,  // (128,36,6) in [0,25)
                    const int*   __restrict__ idx1,  // (128,6,6)  in [0,36)
                    const int*   __restrict__ idx2,  // (128,1,6)  in [0,6)
                    const float* __restrict__ sig0,  // (128,36,64)
                    const float* __restrict__ sig1,  // (128,6,64)
                    const float* __restrict__ sig2,  // (128,1,64)
                    float*       __restrict__ out) { // (64,128,4,4)
  const int t    = blockIdx.x;       // tree
  const int b    = blockIdx.y;       // image
  const int lane = threadIdx.x;      // 0..31
  const int pix  = lane & 15;        // output position oy*4+ox
  const bool hi  = lane >= 16;

  __shared__ float xl[64];           // this image's 8x8 pixels
  __shared__ float h0[16 * kM0];     // layer0 outputs per pixel
  __shared__ float h1[16 * kM1];     // layer1 outputs per pixel

  xl[lane]      = x[b * 64 + lane];
  xl[lane + 32] = x[b * 64 + lane + 32];
  __syncthreads();                   // single-wave WG: lowers to s_nop + waits

  const int oy = pix >> 2, ox = pix & 3;
  const int pbase = oy * 8 + ox;

  // ---- layer 0: 36 nodes, gather from 25 unfolded features ----
  for (int m = 0; m < kM0; ++m) {
    const int* ip = idx0 + (t * kM0 + m) * 6;   // uniform -> scalar loads
    float g[6];
#pragma unroll
    for (int k = 0; k < 6; ++k) {
      int f = ip[k];                            // f = i*5 + j, i,j in [0,5)
      int off = pbase + (f / 5) * 8 + (f % 5);  // addr in 8x8 image tile
      g[k] = xl[off];
    }
    float h = lut_node(g, sig0 + (t * kM0 + m) * 64, lane);
    // lanes l and l^16 store the identical value to the same address:
    // deterministic, and avoids a per-node exec-mask branch.
    h0[pix * kM0 + m] = h;
  }
  __syncthreads();

  // ---- layer 1: 6 nodes, gather from the tree's 36 h0 values ----
  for (int m = 0; m < kM1; ++m) {
    const int* ip = idx1 + (t * kM1 + m) * 6;
    float g[6];
#pragma unroll
    for (int k = 0; k < 6; ++k) g[k] = h0[pix * kM0 + ip[k]];
    float h = lut_node(g, sig1 + (t * kM1 + m) * 64, lane);
    h1[pix * kM1 + m] = h;
  }
  __syncthreads();

  // ---- layer 2: 1 node, gather from the tree's 6 h1 values ----
  {
    const int* ip = idx2 + t * 6;
    float g[6];
#pragma unroll
    for (int k = 0; k < 6; ++k) g[k] = h1[pix * kM1 + ip[k]];
    float h = lut_node(g, sig2 + t * 64, lane);
    // out[b, t, oy, ox] -- one guarded store per block
    if (!hi) out[(b * kT + t) * 16 + pix] = h;
  }
}

// ---------------------------------------------------------------------------
extern "C" void kernel_launch(void* const* d_in, const int* in_sizes, int n_in,
                              void* d_out, int out_size, void* d_ws, size_t ws_size,
                              hipStream_t stream) {
  const float* x    = (const float*)d_in[0];
  const int*   idx0 = (const int*)  d_in[1];
  const float* lut0 = (const float*)d_in[2];
  const int*   idx1 = (const int*)  d_in[3];
  const float* lut1 = (const float*)d_in[4];
  const int*   idx2 = (const int*)  d_in[5];
  const float* lut2 = (const float*)d_in[6];
  float* out = (float*)d_out;
  float* ws  = (float*)d_ws;        // sig0 | sig1 | sig2  (1.41 MB)

  const int total = kN0 + kN1 + kN2;
  sigmoid_pre_kernel<<<(total + 255) / 256, 256, 0, stream>>>(lut0, lut1, lut2, ws);

  dim3 grid(kT, 64);                // (tree, image) -> 8192 wave32 blocks
  lutconv_kernel<<<grid, 32, 0, stream>>>(x, idx0, idx1, idx2,
                                          ws, ws + kN0, ws + kN0 + kN1, out);
}